// SimpleMamba2Block_27685359190820
// MI455X (gfx1250) — compile-verified
//
#include <hip/hip_runtime.h>
#include <math.h>

#define D_MODEL 1024
#define D_STATE 16
#define D_CONV  4
#define BATCH   2
#define SEQ     2048
#define ROWS    (BATCH * SEQ) // 4096

typedef float v2f __attribute__((ext_vector_type(2)));
typedef float v8f __attribute__((ext_vector_type(8)));
typedef int   v4i __attribute__((ext_vector_type(4)));

__device__ __forceinline__ float siluf(float x) { return x / (1.0f + __expf(-x)); }
__device__ __forceinline__ float softplusf(float x) {
    return (x > 20.0f) ? x : log1pf(__expf(x));
}

// CDNA5 async global->LDS path (ASYNCcnt-tracked, bypasses VGPRs). Guarded so
// the file still compiles (with a vectorized synchronous fallback) if this
// toolchain doesn't declare the builtins. Probe-confirmed: builtin exists on
// this toolchain and takes (v4i* src, v4i* dst, imm offset, imm cpol).
#if defined(__AMDGCN__) && \
    __has_builtin(__builtin_amdgcn_global_load_async_to_lds_b128) && \
    __has_builtin(__builtin_amdgcn_s_wait_asynccnt)
#define USE_ASYNC_LDS 1
#else
#define USE_ASYNC_LDS 0
#endif

// ---------------------------------------------------------------------------
// Generic row-major f32 GEMM: C[M,N] = A[M,K] * B[K,N], built on
// v_wmma_f32_16x16x4_f32. Each wave owns one 16x16 C tile; a block of
// (BLOCK/32) waves covers 16 rows x (16*waves) cols. The 16xTK A tile is
// double-buffered in LDS and filled with global_load_async_to_lds_b128 so the
// next tile's fill overlaps the current tile's WMMA chain. Row stride padded
// to 68 floats (272 B): 16B-aligned for B128 writes, bank-conflict-free for
// the column-wise A-fragment reads (bank = (4r + c) mod 64).
// mode==1 applies softplus(acc + bias[col]) at the store (for dt).
// ---------------------------------------------------------------------------
#define TK  64
#define TKP 68

template <int BLOCK>
__global__ __launch_bounds__(BLOCK) void gemm_wmma_f32(
    const float* __restrict__ A, const float* __restrict__ B,
    float* __restrict__ C, int K, int lda, int ldb, int ldc,
    const float* __restrict__ bias, int mode)
{
    constexpr int CH = (16 * TK / 4) / BLOCK; // float4 chunks per thread per tile
    __shared__ __align__(16) float As[2][16][TKP];

    const int tid  = threadIdx.x;
    const int lane = tid & 31;
    const int wave = tid >> 5;
    const int l16  = lane & 15;
    const int half = lane >> 4;          // 0: K pair {0,1}, 1: K pair {2,3}

    const int m0 = blockIdx.y * 16;
    const int n0 = (blockIdx.x * (BLOCK >> 5) + wave) * 16;

    // Stage one 16 x TK A tile into LDS buffer `buf`.
    auto issue_tile = [&](int k0, int buf) {
        #pragma unroll
        for (int j = 0; j < CH; ++j) {
            const int q  = tid + BLOCK * j;     // chunk id, 0..255
            const int r  = q >> 4;              // row 0..15
            const int c4 = (q & 15) << 2;       // col 0,4,...,60
            const float* gp = A + (size_t)(m0 + r) * lda + (k0 + c4);
            float* lp = &As[buf][r][c4];
#if USE_ASYNC_LDS
            __builtin_amdgcn_global_load_async_to_lds_b128(
                (v4i*)(void*)gp, (v4i*)(void*)lp, 0, 0);
#else
            *(float4*)lp = *(const float4*)gp;
#endif
        }
    };

    v8f acc = {};
    const int ntiles = K / TK;

    issue_tile(0, 0);
    for (int ti = 0; ti < ntiles; ++ti) {
        const int buf = ti & 1;
        if (ti + 1 < ntiles) {
            issue_tile((ti + 1) * TK, buf ^ 1);
#if USE_ASYNC_LDS
            // current tile complete; next tile's CH ops still in flight
            __builtin_amdgcn_s_wait_asynccnt(CH);
#endif
        } else {
#if USE_ASYNC_LDS
            __builtin_amdgcn_s_wait_asynccnt(0);
#endif
        }
        __syncthreads();

        const int k0 = ti * TK;
        #pragma unroll
        for (int kk = 0; kk < TK; kk += 4) {
            // A fragment (16x4): lanes 0-15 -> K = kk,kk+1 ; lanes 16-31 -> K = kk+2,kk+3
            v2f af, bfrag;
            af.x = As[buf][l16][kk + 2 * half + 0];
            af.y = As[buf][l16][kk + 2 * half + 1];
            // B fragment (4x16), mirrored layout, loaded straight from global (L2-resident)
            const float* bp = B + (size_t)(k0 + kk + 2 * half) * ldb + (n0 + l16);
            bfrag.x = bp[0];
            bfrag.y = bp[ldb];
            acc = __builtin_amdgcn_wmma_f32_16x16x4_f32(
                false, af, false, bfrag, (short)0, acc, false, false);
        }
        __syncthreads();
    }

    const int col = n0 + l16;
    const float bv = (mode == 1) ? bias[col] : 0.0f;
    #pragma unroll
    for (int i = 0; i < 8; ++i) {
        const int row = m0 + i + 8 * half;   // C/D layout: M=i (lanes 0-15), M=i+8 (lanes 16-31)
        float v = acc[i];
        if (mode == 1) v = softplusf(v + bv);
        C[(size_t)row * ldc + col] = v;
    }
}

// ---------------------------------------------------------------------------
// Depthwise causal conv (width 4) + bias + SiLU.
// Reads x_ssm = xz[:, 0:D_MODEL] (row stride 2*D_MODEL), writes x_act.
// ---------------------------------------------------------------------------
__global__ __launch_bounds__(256) void conv_silu_kernel(
    const float* __restrict__ xz, const float* __restrict__ conv_w,
    const float* __restrict__ conv_b, float* __restrict__ xact)
{
    const int idx = blockIdx.x * blockDim.x + threadIdx.x; // over ROWS*D_MODEL
    const int d   = idx & (D_MODEL - 1);
    const int row = idx >> 10;        // / D_MODEL
    const int t   = row & (SEQ - 1);
    const int b   = row >> 11;        // / SEQ

    float s = conv_b[d];
    #pragma unroll
    for (int j = 0; j < D_CONV; ++j) {
        const int tt = t + j - (D_CONV - 1);
        if (tt >= 0)
            s += xz[(size_t)(b * SEQ + tt) * (2 * D_MODEL) + d] * conv_w[d * D_CONV + j];
    }
    xact[idx] = siluf(s);
}

// ---------------------------------------------------------------------------
// Selective scan. One thread per (batch, channel); h[16] in VGPRs.
// B_t / C_t (32 floats, shared across all channels) broadcast via LDS.
// Fused epilogue: ypre = (y + x*D) * silu(z).
// ---------------------------------------------------------------------------
__global__ __launch_bounds__(256) void scan_kernel(
    const float* __restrict__ xact, const float* __restrict__ dt,
    const float* __restrict__ bc,   const float* __restrict__ xz,
    const float* __restrict__ A_log, const float* __restrict__ Dp,
    float* __restrict__ ypre)
{
    __shared__ float bcs[2 * D_STATE];

    const int d = blockIdx.y * blockDim.x + threadIdx.x;
    const int b = blockIdx.x;
    const float Aneg = -__expf(A_log[d]);
    const float Dd   = Dp[d];

    float h[D_STATE];
    #pragma unroll
    for (int n = 0; n < D_STATE; ++n) h[n] = 0.0f;

    for (int t = 0; t < SEQ; ++t) {
        const int row = b * SEQ + t;
        if (threadIdx.x < 2 * D_STATE)
            bcs[threadIdx.x] = bc[(size_t)row * (2 * D_STATE) + threadIdx.x];
        __syncthreads();

        const float xv = xact[(size_t)row * D_MODEL + d];
        const float dA = __expf(dt[(size_t)row * D_MODEL + d] * Aneg);
        float y = 0.0f;
        #pragma unroll
        for (int n = 0; n < D_STATE; ++n) {
            h[n] = h[n] * dA + xv * bcs[n];
            y   += h[n] * bcs[D_STATE + n];
        }
        const float zv = xz[(size_t)row * (2 * D_MODEL) + D_MODEL + d];
        ypre[(size_t)row * D_MODEL + d] = (y + xv * Dd) * siluf(zv);
        __syncthreads();
    }
}

// ---------------------------------------------------------------------------
extern "C" void kernel_launch(void* const* d_in, const int* in_sizes, int n_in,
                              void* d_out, int out_size, void* d_ws, size_t ws_size,
                              hipStream_t stream)
{
    (void)in_sizes; (void)n_in; (void)out_size; (void)ws_size;

    const float* x      = (const float*)d_in[0];
    const float* W_in   = (const float*)d_in[1];
    const float* conv_w = (const float*)d_in[2];
    const float* conv_b = (const float*)d_in[3];
    const float* W_x    = (const float*)d_in[4];
    const float* W_dt   = (const float*)d_in[5];
    const float* b_dt   = (const float*)d_in[6];
    const float* A_log  = (const float*)d_in[7];
    const float* Dp     = (const float*)d_in[8];
    const float* W_out  = (const float*)d_in[9];
    float* out = (float*)d_out;
    float* ws  = (float*)d_ws;

    // workspace layout (floats)
    float* XZ   = ws;                                    // ROWS x 2*D_MODEL
    float* XACT = XZ   + (size_t)ROWS * 2 * D_MODEL;     // ROWS x D_MODEL
    float* DT   = XACT + (size_t)ROWS * D_MODEL;         // ROWS x D_MODEL
    float* BC   = DT   + (size_t)ROWS * D_MODEL;         // ROWS x 2*D_STATE
    float* YPRE = BC   + (size_t)ROWS * 2 * D_STATE;     // ROWS x D_MODEL

    // 1) xz = x @ W_in          (4096x1024 @ 1024x2048)
    gemm_wmma_f32<128><<<dim3((2 * D_MODEL) / 64, ROWS / 16), 128, 0, stream>>>(
        x, W_in, XZ, D_MODEL, D_MODEL, 2 * D_MODEL, 2 * D_MODEL, nullptr, 0);

    // 2) x_act = silu(causal_conv(x_ssm) + conv_b)
    conv_silu_kernel<<<(ROWS * D_MODEL) / 256, 256, 0, stream>>>(
        XZ, conv_w, conv_b, XACT);

    // 3) dt = softplus(x_act @ W_dt + b_dt)   (4096x1024 @ 1024x1024)
    gemm_wmma_f32<128><<<dim3(D_MODEL / 64, ROWS / 16), 128, 0, stream>>>(
        XACT, W_dt, DT, D_MODEL, D_MODEL, D_MODEL, D_MODEL, b_dt, 1);

    // 4) BC = x_act @ W_x        (4096x1024 @ 1024x32) -> 2 waves/block
    gemm_wmma_f32<64><<<dim3(1, ROWS / 16), 64, 0, stream>>>(
        XACT, W_x, BC, D_MODEL, D_MODEL, 2 * D_STATE, 2 * D_STATE, nullptr, 0);

    // 5) selective scan + (y + x*D) * silu(z)
    scan_kernel<<<dim3(BATCH, D_MODEL / 256), 256, 0, stream>>>(
        XACT, DT, BC, XZ, A_log, Dp, YPRE);

    // 6) out = ypre @ W_out      (4096x1024 @ 1024x1024)
    gemm_wmma_f32<128><<<dim3(D_MODEL / 64, ROWS / 16), 128, 0, stream>>>(
        YPRE, W_out, out, D_MODEL, D_MODEL, D_MODEL, D_MODEL, nullptr, 0);
}